// SpatialModule_45896020525700
// MI455X (gfx1250) — compile-verified
//
#include <hip/hip_runtime.h>

// ---------------------------------------------------------------------------
// GraphSAGE (mean) x3 + BN(eval) + ReLU for MI455X (gfx1250, wave32, WMMA)
//   N=50000 nodes, E=800000 edges, C: 128 -> 256 -> 256 -> 256
//
//   1) CSR-by-destination build (1.6M atomics), gather-mean with vector loads
//      (h is L2-resident in the 192MB L2; no RMW serialization).
//   2) Fused dual-GEMM + BN + ReLU with V_WMMA_F32_16X16X4_F32.
//      Each block = one 16-row stripe; its A tiles (agg,h : 2 x 16xC = 32KB)
//      are staged into LDS ONCE by the Tensor Data Mover (tensor_load_to_lds,
//      TENSORcnt), then shared by all 8 column-pair waves -> 8x less A traffic.
// ---------------------------------------------------------------------------

typedef __attribute__((ext_vector_type(2))) float v2f;
typedef __attribute__((ext_vector_type(8))) float v8f;
typedef __attribute__((ext_vector_type(4))) unsigned int v4u;
typedef __attribute__((ext_vector_type(8))) int v8i_t;
typedef __attribute__((ext_vector_type(4))) int v4i_t;

#define NN 50000
#define EE 800000
#define HH 256
#define BN_EPS 1e-5f
#define SCAN_T 1024
#define SCAN_CHUNK 49  // ceil(50000/1024)

// ---------------- zero fill (ints) ----------------
__global__ void fill_zero_i32(int* __restrict__ p, int n) {
  int i = blockIdx.x * blockDim.x + threadIdx.x;
  int st = gridDim.x * blockDim.x;
  for (; i < n; i += st) p[i] = 0;
}

// ---------------- in-degree count (int atomic) ----------------
__global__ void count_deg_kernel(const long long* __restrict__ dst,
                                 int* __restrict__ counts, int E) {
  int e = blockIdx.x * blockDim.x + threadIdx.x;
  if (e < E) {
    int d = (int)dst[e];
    __hip_atomic_fetch_add(counts + d, 1, __ATOMIC_RELAXED,
                           __HIP_MEMORY_SCOPE_AGENT);
  }
}

// ---------------- single-block exclusive scan -> rowptr, cursor ----------------
__global__ __launch_bounds__(SCAN_T) void scan_kernel(
    const int* __restrict__ counts, int* __restrict__ rowptr,
    int* __restrict__ cursor) {
  __shared__ int partial[SCAN_T];
  int t = threadIdx.x;
  int base = t * SCAN_CHUNK;
  int sum = 0;
#pragma unroll 4
  for (int i = 0; i < SCAN_CHUNK; ++i) {
    int idx = base + i;
    if (idx < NN) sum += counts[idx];
  }
  partial[t] = sum;
  __syncthreads();
  for (int off = 1; off < SCAN_T; off <<= 1) {  // Hillis-Steele in LDS
    int v = 0;
    if (t >= off) v = partial[t - off];
    __syncthreads();
    if (t >= off) partial[t] += v;
    __syncthreads();
  }
  int run = (t == 0) ? 0 : partial[t - 1];
#pragma unroll 4
  for (int i = 0; i < SCAN_CHUNK; ++i) {
    int idx = base + i;
    if (idx < NN) {
      rowptr[idx] = run;
      cursor[idx] = run;
      run += counts[idx];
    }
  }
  if (t == SCAN_T - 1) rowptr[NN] = partial[SCAN_T - 1];  // == E
}

// ---------------- CSR fill: srclist bucketed by dst ----------------
__global__ void fill_csr_kernel(const long long* __restrict__ src,
                                const long long* __restrict__ dst,
                                int* __restrict__ cursor,
                                int* __restrict__ srclist, int E) {
  int e = blockIdx.x * blockDim.x + threadIdx.x;
  if (e < E) {
    int d = (int)dst[e];
    int pos = __hip_atomic_fetch_add(cursor + d, 1, __ATOMIC_RELAXED,
                                     __HIP_MEMORY_SCOPE_AGENT);
    srclist[pos] = (int)src[e];
  }
}

// ---------------- gather-mean: agg[d] = mean_{s in N(d)} h[s] ----------------
template <int C>
__global__ __launch_bounds__(256) void gather_mean_kernel(
    const int* __restrict__ rowptr, const int* __restrict__ srclist,
    const float* __restrict__ h, float* __restrict__ agg) {
  int lane = threadIdx.x & 31;
  int node = (blockIdx.x * (int)blockDim.x + threadIdx.x) >> 5;
  if (node >= NN) return;
  int start = rowptr[node];
  int end = rowptr[node + 1];
  float4 acc0 = {0.f, 0.f, 0.f, 0.f};
  float4 acc1 = {0.f, 0.f, 0.f, 0.f};
  for (int e = start; e < end; e += 32) {
    int nb = end - e;
    if (nb > 32) nb = 32;
    int sidx = (lane < nb) ? srclist[e + lane] : 0;
    for (int j = 0; j < nb; ++j) {
      int s = __shfl(sidx, j);
      const float4* hs = (const float4*)(h + (size_t)s * C);
      float4 v0 = hs[lane];
      acc0.x += v0.x; acc0.y += v0.y; acc0.z += v0.z; acc0.w += v0.w;
      if (C == 256) {
        float4 v1 = hs[lane + 32];
        acc1.x += v1.x; acc1.y += v1.y; acc1.z += v1.z; acc1.w += v1.w;
      }
    }
  }
  int degi = end - start;
  float inv = 1.0f / (float)(degi > 0 ? degi : 1);
  float4* ao = (float4*)(agg + (size_t)node * C);
  float4 o0 = {acc0.x * inv, acc0.y * inv, acc0.z * inv, acc0.w * inv};
  ao[lane] = o0;
  if (C == 256) {
    float4 o1 = {acc1.x * inv, acc1.y * inv, acc1.z * inv, acc1.w * inv};
    ao[lane + 32] = o1;
  }
}

// ---------------- TDM: 2D tile (rows x C f32, row stride C) -> LDS ----------
// D# per CDNA5 ISA ch.8: group0 = {count/lds_addr/global_addr/type},
// group1 = {data_size, tensor dims/strides, tile dims}; groups 2/3 unused (0).
// amdgpu-toolchain builtin form: (v4u, v8i, v4i, v4i, v8i, i32 cpol)
__device__ __forceinline__ void tdm_load_tile_f32(const float* gsrc,
                                                  unsigned lds_off, int C,
                                                  int rows) {
  unsigned long long ga = (unsigned long long)(uintptr_t)gsrc;
  v4u g0;
  g0[0] = 1u;                                            // count=1, user mode
  g0[1] = lds_off;                                       // lds_addr (bytes)
  g0[2] = (unsigned)(ga & 0xFFFFFFFFu);                  // global_addr[31:0]
  g0[3] = (unsigned)((ga >> 32) & 0x01FFFFFFu) | (2u << 30);  // [56:32]|type=2
  v8i_t g1;
  g1[0] = 2 << 16;                                       // data_size=4B, mask=0
  g1[1] = (C & 0xFFFF) << 16;                            // tensor_dim0 lo16
  g1[2] = ((unsigned)C >> 16) | ((NN & 0xFFFF) << 16);   // dim0 hi | dim1 lo
  g1[3] = (NN >> 16) | ((C & 0xFFFF) << 16);             // dim1 hi | tile_dim0
  g1[4] = rows & 0xFFFF;                                 // tile_dim1; tile_dim2=0
  g1[5] = C;                                             // tensor_dim0_stride lo32
  g1[6] = 0;                                             // stride hi | dim1_stride lo
  g1[7] = 0;
  v4i_t gz4 = {0, 0, 0, 0};
  v8i_t gz8 = {0, 0, 0, 0, 0, 0, 0, 0};
  __builtin_amdgcn_tensor_load_to_lds(g0, g1, gz4, gz4, gz8, 0);
}

// ---------------- fused dual-GEMM + bias + BN + ReLU ----------------
// block = 8 waves = all 8 column-tile-pairs of one 16-row stripe.
// A tiles staged in LDS by TDM, shared by all waves; B streamed from L2.
template <int C>
__global__ __launch_bounds__(256) void sage_gemm_bn_relu_kernel(
    const float* __restrict__ agg, const float* __restrict__ hin,
    const float* __restrict__ Wl, const float* __restrict__ Wr,
    const float* __restrict__ bias,
    const float* __restrict__ g, const float* __restrict__ bt,
    const float* __restrict__ rm, const float* __restrict__ rv,
    float* __restrict__ out) {
  __shared__ float sA[16 * C];
  __shared__ float sH[16 * C];

  int tm = blockIdx.x;              // row tile (3125 blocks, exact)
  int tp = threadIdx.x >> 5;        // column-tile pair = wave id (0..7)
  int lane = threadIdx.x & 31;

  if (tp == 0) {                    // one wave issues both TDM descriptors
    tdm_load_tile_f32(agg + (size_t)tm * 16 * C,
                      (unsigned)(uintptr_t)&sA[0], C, 16);
    tdm_load_tile_f32(hin + (size_t)tm * 16 * C,
                      (unsigned)(uintptr_t)&sH[0], C, 16);
    __builtin_amdgcn_s_wait_tensorcnt(0);
  }
  __syncthreads();

  int rlow = lane & 15;
  int khalf = lane >> 4;            // 0: K=k,k+1  1: K=k+2,k+3
  int col0 = tp * 32 + rlow;
  int col1 = col0 + 16;

  const float* __restrict__ Arow = sA + rlow * C;   // LDS, ds_load_b64
  const float* __restrict__ Hrow = sH + rlow * C;

  v8f acc0 = {0.f, 0.f, 0.f, 0.f, 0.f, 0.f, 0.f, 0.f};
  v8f acc1 = {0.f, 0.f, 0.f, 0.f, 0.f, 0.f, 0.f, 0.f};

#pragma unroll 8
  for (int k = 0; k < C; k += 4) {
    int kb = k + 2 * khalf;
    float2 ta = *(const float2*)(Arow + kb);
    float2 th = *(const float2*)(Hrow + kb);
    v2f aA = {ta.x, ta.y};
    v2f aH = {th.x, th.y};
    const float* wlk = Wl + (size_t)kb * HH;   // [C, H] row-major
    const float* wrk = Wr + (size_t)kb * HH;
    v2f bL0 = {wlk[col0], wlk[HH + col0]};
    v2f bL1 = {wlk[col1], wlk[HH + col1]};
    v2f bR0 = {wrk[col0], wrk[HH + col0]};
    v2f bR1 = {wrk[col1], wrk[HH + col1]};
    acc0 = __builtin_amdgcn_wmma_f32_16x16x4_f32(false, aA, false, bL0,
                                                 (short)0, acc0, false, false);
    acc0 = __builtin_amdgcn_wmma_f32_16x16x4_f32(false, aH, false, bR0,
                                                 (short)0, acc0, false, false);
    acc1 = __builtin_amdgcn_wmma_f32_16x16x4_f32(false, aA, false, bL1,
                                                 (short)0, acc1, false, false);
    acc1 = __builtin_amdgcn_wmma_f32_16x16x4_f32(false, aH, false, bR1,
                                                 (short)0, acc1, false, false);
  }

  // epilogue: (acc + bias - rm)*g*rsqrt(rv+eps) + bt, ReLU
  int rbase = tm * 16 + khalf * 8;
  {
    float scale = g[col0] * rsqrtf(rv[col0] + BN_EPS);
    float shift = bt[col0] + (bias[col0] - rm[col0]) * scale;
#pragma unroll
    for (int r = 0; r < 8; ++r) {
      float o = acc0[r] * scale + shift;
      out[(size_t)(rbase + r) * HH + col0] = fmaxf(o, 0.0f);
    }
  }
  {
    float scale = g[col1] * rsqrtf(rv[col1] + BN_EPS);
    float shift = bt[col1] + (bias[col1] - rm[col1]) * scale;
#pragma unroll
    for (int r = 0; r < 8; ++r) {
      float o = acc1[r] * scale + shift;
      out[(size_t)(rbase + r) * HH + col1] = fmaxf(o, 0.0f);
    }
  }
}

// ---------------------------------------------------------------------------
extern "C" void kernel_launch(void* const* d_in, const int* in_sizes, int n_in,
                              void* d_out, int out_size, void* d_ws,
                              size_t ws_size, hipStream_t stream) {
  // order: x, edge_index, gamma, beta, rmean, rvar, Wl0,Wr0,b0, Wl1,Wr1,b1, Wl2,Wr2,b2
  const float*     x     = (const float*)d_in[0];
  const long long* ei    = (const long long*)d_in[1];   // int64 [2, E]
  const float*     gamma = (const float*)d_in[2];
  const float*     beta  = (const float*)d_in[3];
  const float*     rmean = (const float*)d_in[4];
  const float*     rvar  = (const float*)d_in[5];
  const float*     Wl0 = (const float*)d_in[6];
  const float*     Wr0 = (const float*)d_in[7];
  const float*     b0  = (const float*)d_in[8];
  const float*     Wl1 = (const float*)d_in[9];
  const float*     Wr1 = (const float*)d_in[10];
  const float*     b1  = (const float*)d_in[11];
  const float*     Wl2 = (const float*)d_in[12];
  const float*     Wr2 = (const float*)d_in[13];
  const float*     b2  = (const float*)d_in[14];

  const long long* srcI = ei;
  const long long* dstI = ei + EE;

  int* wsi      = (int*)d_ws;
  int* counts   = wsi;                        // [N]
  int* rowptr   = counts + NN;                // [N+1]
  int* cursor   = rowptr + NN + 1;            // [N]
  int* srclist  = cursor + NN;                // [E]
  size_t ihdr   = (size_t)(NN + (NN + 1) + NN + EE);
  ihdr          = (ihdr + 3) & ~(size_t)3;    // 16B alignment
  float* agg    = (float*)(wsi + ihdr);       // [N*256]
  float* hA     = agg + (size_t)NN * HH;      // [N*256]
  float* hB     = hA + (size_t)NN * HH;       // [N*256]
  float* outF   = (float*)d_out;

  const int gatherBlocks = (NN * 32) / 256;   // one wave per node
  const int gemmBlocks   = NN / 16;           // one block per 16-row stripe

  // ---- CSR-by-destination build (edges constant across layers) ----
  fill_zero_i32<<<256, 256, 0, stream>>>(counts, NN);
  count_deg_kernel<<<(EE + 255) / 256, 256, 0, stream>>>(dstI, counts, EE);
  scan_kernel<<<1, SCAN_T, 0, stream>>>(counts, rowptr, cursor);
  fill_csr_kernel<<<(EE + 255) / 256, 256, 0, stream>>>(srcI, dstI, cursor,
                                                        srclist, EE);

  // ---- layer 0 (C = 128): x -> hA ----
  gather_mean_kernel<128><<<gatherBlocks, 256, 0, stream>>>(rowptr, srclist, x, agg);
  sage_gemm_bn_relu_kernel<128><<<gemmBlocks, 256, 0, stream>>>(
      agg, x, Wl0, Wr0, b0, gamma + 0 * HH, beta + 0 * HH,
      rmean + 0 * HH, rvar + 0 * HH, hA);

  // ---- layer 1 (C = 256): hA -> hB ----
  gather_mean_kernel<256><<<gatherBlocks, 256, 0, stream>>>(rowptr, srclist, hA, agg);
  sage_gemm_bn_relu_kernel<256><<<gemmBlocks, 256, 0, stream>>>(
      agg, hA, Wl1, Wr1, b1, gamma + 1 * HH, beta + 1 * HH,
      rmean + 1 * HH, rvar + 1 * HH, hB);

  // ---- layer 2 (C = 256): hB -> d_out ----
  gather_mean_kernel<256><<<gatherBlocks, 256, 0, stream>>>(rowptr, srclist, hB, agg);
  sage_gemm_bn_relu_kernel<256><<<gemmBlocks, 256, 0, stream>>>(
      agg, hB, Wl2, Wr2, b2, gamma + 2 * HH, beta + 2 * HH,
      rmean + 2 * HH, rvar + 2 * HH, outF);
}